// BADDescriptor_30374008717910
// MI455X (gfx1250) — compile-verified
//
#include <hip/hip_runtime.h>
#include <stdint.h>

#define IMG_H 480
#define IMG_W 640
#define HW    (IMG_H * IMG_W)
#define NPAIR 256

// ---------------------------------------------------------------------------
// Pass 1: horizontal box sums for radii 1,2,3 (edge-clamped), computed
// incrementally (7 reads -> 3 outputs per pixel). Image is 1.2MB -> L2 hits.
// ---------------------------------------------------------------------------
__global__ __launch_bounds__(256) void hsum_kernel(const float* __restrict__ x,
                                                   float* __restrict__ hsum) {
    int idx = blockIdx.x * 256 + threadIdx.x;
    if (idx >= HW) return;
    int y  = idx / IMG_W;
    int xx = idx - y * IMG_W;
    const float* row = x + (size_t)y * IMG_W;
    float c  = row[xx];
    float l1 = row[max(xx - 1, 0)],        r1 = row[min(xx + 1, IMG_W - 1)];
    float l2 = row[max(xx - 2, 0)],        r2 = row[min(xx + 2, IMG_W - 1)];
    float l3 = row[max(xx - 3, 0)],        r3 = row[min(xx + 3, IMG_W - 1)];
    float s1 = l1 + c + r1;
    float s2 = s1 + l2 + r2;
    float s3 = s2 + l3 + r3;
    hsum[idx]          = s1;
    hsum[HW + idx]     = s2;
    hsum[2 * HW + idx] = s3;
}

// ---------------------------------------------------------------------------
// Pass 2: vertical box sums (edge-clamped) + divide by area -> mean maps
// M_r(y,x), r = 1..3.  3 x 480 x 640 x 4B = 3.7MB, stays in L2.
// ---------------------------------------------------------------------------
__global__ __launch_bounds__(256) void vmean_kernel(const float* __restrict__ hsum,
                                                    float* __restrict__ mean) {
    int idx = blockIdx.x * 256 + threadIdx.x;
    if (idx >= HW) return;
    int y  = idx / IMG_W;
    int xx = idx - y * IMG_W;
    const float inv_area[3] = {1.0f / 9.0f, 1.0f / 25.0f, 1.0f / 49.0f};
#pragma unroll
    for (int r = 1; r <= 3; ++r) {
        const float* m = hsum + (size_t)(r - 1) * HW;
        float s = 0.0f;
#pragma unroll
        for (int d = -3; d <= 3; ++d) {
            if (d < -r || d > r) continue;
            int yy = min(max(y + d, 0), IMG_H - 1);
            s += m[(size_t)yy * IMG_W + xx];
        }
        mean[(size_t)(r - 1) * HW + idx] = s * inv_area[r - 1];
    }
}

// ---------------------------------------------------------------------------
// Pass 2b: pack per-pair constants once:  {lds_base1, lds_base2, thr_bits}
// so the hot loop needs a single uniform s_load_b128 per pair.
// ---------------------------------------------------------------------------
__global__ __launch_bounds__(NPAIR) void pair_params_kernel(
    const float* __restrict__ off_x1, const float* __restrict__ off_x2,
    const float* __restrict__ off_y1, const float* __restrict__ off_y2,
    const int*   __restrict__ radii,  const float* __restrict__ thresholds,
    int4* __restrict__ params) {
    int p = threadIdx.x;
    int r  = radii[p];
    int wb = (r - 1) << 12;
    int b1 = wb + ((int)off_y1[p] + 16) * 64 + ((int)off_x1[p] + 16);
    int b2 = wb + ((int)off_y2[p] + 16) * 64 + ((int)off_x2[p] + 16);
    params[p] = make_int4(b1, b2, __float_as_int(thresholds[p]), 0);
}

// ---------------------------------------------------------------------------
// Pass 3 (the 99% pass): per 32x32 output tile, async-DMA the 3 x 64 x 64
// clamped gather window of the mean maps into LDS (48KB, ASYNCcnt path),
// then for all 256 pairs emit  M_r(c1) - M_r(c2) - thr  with 2 bank-conflict-
// free ds_loads per output and non-temporal coalesced stores (output is
// 315MB > 192MB L2; NT keeps the mean maps resident).
// ---------------------------------------------------------------------------
__global__ __launch_bounds__(256) void bad_kernel(
    const float* __restrict__ mean,
    const int4*  __restrict__ params,
    float* __restrict__ out) {
    __shared__ float smem[3 * 64 * 64];   // 49152 bytes -> 6 blocks / WGP

    const int tid = threadIdx.x;
    const int tx  = tid & 31;
    const int ty0 = tid >> 5;             // 0..7
    const int X0  = blockIdx.x << 5;
    const int Y0  = blockIdx.y << 5;

    // ---- Stage window via CDNA5 async global->LDS DMA ----
    // LDS plane r, row i, col j gets M_r at clamped (Y0-16+i, X0-16+j), for i,j in [0,64)
#pragma unroll 4
    for (int k = 0; k < 48; ++k) {
        int e   = tid + (k << 8);          // 0 .. 12287
        int r   = e >> 12;                 // radius plane 0..2
        int rem = e & 4095;
        int i   = rem >> 6;
        int j   = rem & 63;
        int gy  = min(max(Y0 - 16 + i, 0), IMG_H - 1);
        int gx  = min(max(X0 - 16 + j, 0), IMG_W - 1);
        const float* gp = mean + (size_t)r * HW + (size_t)gy * IMG_W + gx;
        // Low 32 bits of a generic LDS pointer == LDS byte address (ISA 10.2).
        unsigned lds_addr = (unsigned)(unsigned long long)(uintptr_t)&smem[e];
        asm volatile("global_load_async_to_lds_b32 %0, %1, off"
                     :: "v"(lds_addr), "v"(gp)
                     : "memory");
    }
#if __has_builtin(__builtin_amdgcn_s_wait_asynccnt)
    __builtin_amdgcn_s_wait_asynccnt(0);
#else
    asm volatile("s_wait_asynccnt 0" ::: "memory");
#endif
    __syncthreads();

    // ---- Main loop: 256 pairs x 4 pixels/thread ----
    const int pixbase = ty0 * 64 + tx;     // tile-local (ty,tx) in LDS units
    float* outbase = out + (size_t)(Y0 + ty0) * IMG_W + X0 + tx;

#pragma unroll 2
    for (int p = 0; p < NPAIR; ++p) {
        const int4  pp    = params[p];               // uniform -> s_load_b128
        const int   base1 = pp.x;
        const int   base2 = pp.y;
        const float thr   = __int_as_float(pp.z);
        float* po = outbase + (size_t)p * HW;
#pragma unroll
        for (int k = 0; k < 4; ++k) {
            const int pix = pixbase + k * (8 * 64);
            const float sa = smem[base1 + pix];
            const float sb = smem[base2 + pix];
            __builtin_nontemporal_store(sa - sb - thr, po + (size_t)k * (8 * IMG_W));
        }
    }
}

// ---------------------------------------------------------------------------
extern "C" void kernel_launch(void* const* d_in, const int* in_sizes, int n_in,
                              void* d_out, int out_size, void* d_ws, size_t ws_size,
                              hipStream_t stream) {
    const float* x       = (const float*)d_in[0];
    const float* off_x1  = (const float*)d_in[1];
    const float* off_x2  = (const float*)d_in[2];
    const float* off_y1  = (const float*)d_in[3];
    const float* off_y2  = (const float*)d_in[4];
    const int*   radii   = (const int*)  d_in[5];
    const float* thresh  = (const float*)d_in[6];
    float* out  = (float*)d_out;

    float* hsum   = (float*)d_ws;                 // 3*HW floats
    float* mean   = hsum + 3 * HW;                // 3*HW floats
    int4*  params = (int4*)(mean + 3 * HW);       // 256 * 16B (total < 7.4MB)

    const int blocks = (HW + 255) / 256;
    hsum_kernel <<<blocks, 256, 0, stream>>>(x, hsum);
    vmean_kernel<<<blocks, 256, 0, stream>>>(hsum, mean);
    pair_params_kernel<<<1, NPAIR, 0, stream>>>(off_x1, off_x2, off_y1, off_y2,
                                                radii, thresh, params);

    dim3 grid(IMG_W / 32, IMG_H / 32);            // 20 x 15 tiles
    bad_kernel<<<grid, 256, 0, stream>>>(mean, params, out);
}